// ReactionGAT_62732292326005
// MI455X (gfx1250) — compile-verified
//
#include <hip/hip_runtime.h>
#include <math.h>

typedef __attribute__((ext_vector_type(2))) float v2f;
typedef __attribute__((ext_vector_type(8))) float v8f;

#define HIDF 128

// ---------------------------------------------------------------------------
// utility: float atomic max via ordered-int trick (init must be -inf)
// ---------------------------------------------------------------------------
__device__ __forceinline__ void atomicMaxFloat(float* addr, float value) {
  if (value >= 0.0f) {
    atomicMax((int*)addr, __float_as_int(value));
  } else {
    atomicMin((unsigned int*)addr, __float_as_uint(value));
  }
}

__global__ void fill_kernel(float* p, int n, float v) {
  int i = blockIdx.x * blockDim.x + threadIdx.x;
  if (i < n) p[i] = v;
}

// zero-pad x[N,FIN] -> xpad[N,KP] (KP = FIN rounded up to multiple of 4)
__global__ void pad_kernel(const float* __restrict__ x, float* __restrict__ xpad,
                           int N, int FIN, int KP) {
  int i = blockIdx.x * blockDim.x + threadIdx.x;
  if (i >= N * KP) return;
  int n = i / KP;
  int c = i - n * KP;
  xpad[i] = (c < FIN) ? x[(size_t)n * FIN + c] : 0.0f;
}

// ---------------------------------------------------------------------------
// GEMM: out[N,128] = A[N,K] @ W[K,128] using V_WMMA_F32_16X16X4_F32.
// Requires K % 4 == 0 (pad A beforehand). One wave computes a 16x64 strip:
// 4 accumulator fragments share one A fragment per K-step.
// A fragment (32-bit 16x4): lane m = lane&15, K-pair = 2*(lane>>4)  -> v2f.
// C fragment: VGPR j -> row j + 8*(lane>>4), col = lane&15.
// ---------------------------------------------------------------------------
__global__ void gemm_wmma(const float* __restrict__ A, const float* __restrict__ W,
                          float* __restrict__ out, int N, int K, int tiles) {
  int wave = threadIdx.x >> 5;
  int tile = blockIdx.x * 4 + wave;
  if (tile >= tiles) return;               // wave-uniform exit, EXEC stays full
  int rt = tile >> 1;                      // 2 col-groups of 64 across HID=128
  int cg = tile & 1;
  int row0 = rt * 16;
  int col0 = cg * 64;
  int lane = threadIdx.x & 31;
  int mr   = lane & 15;                    // A row / B,C col
  int koff = (lane >> 4) * 2;              // 0 or 2

  const float* arow = A + (size_t)(row0 + mr) * K + koff;
  const float* wcol = W + col0 + mr;

  v8f acc0 = {}, acc1 = {}, acc2 = {}, acc3 = {};
  for (int k = 0; k < K; k += 4) {
    v2f av = *(const v2f*)(arow + k);
    const float* w0 = wcol + (size_t)(k + koff) * HIDF;
    const float* w1 = w0 + HIDF;
    v2f bv0 = {w0[0],  w1[0]};
    v2f bv1 = {w0[16], w1[16]};
    v2f bv2 = {w0[32], w1[32]};
    v2f bv3 = {w0[48], w1[48]};
    acc0 = __builtin_amdgcn_wmma_f32_16x16x4_f32(false, av, false, bv0, (short)0, acc0, false, false);
    acc1 = __builtin_amdgcn_wmma_f32_16x16x4_f32(false, av, false, bv1, (short)0, acc1, false, false);
    acc2 = __builtin_amdgcn_wmma_f32_16x16x4_f32(false, av, false, bv2, (short)0, acc2, false, false);
    acc3 = __builtin_amdgcn_wmma_f32_16x16x4_f32(false, av, false, bv3, (short)0, acc3, false, false);
  }

  int mbase = (lane >> 4) * 8;
  float* obase = out + (size_t)(row0 + mbase) * HIDF + col0 + mr;
#pragma unroll
  for (int j = 0; j < 8; ++j) {
    float* orow = obase + (size_t)j * HIDF;
    orow[0]  = acc0[j];
    orow[16] = acc1[j];
    orow[32] = acc2[j];
    orow[48] = acc3[j];
  }
}

// ---------------------------------------------------------------------------
// per-node attention scores: als[n,h] = sum_c h[n,h*C+c] * a_s[h*C+c]
// ---------------------------------------------------------------------------
__global__ void scores_kernel(const float* __restrict__ h, const float* __restrict__ a_s,
                              const float* __restrict__ a_d, float* __restrict__ als,
                              float* __restrict__ ald, int N, int H) {
  int n = blockIdx.x * blockDim.x + threadIdx.x;
  if (n >= N) return;
  int C = HIDF / H;
  const float* hn = h + (size_t)n * HIDF;
  for (int hh = 0; hh < H; ++hh) {
    float ss = 0.f, sd = 0.f;
    for (int c = 0; c < C; ++c) {
      float v = hn[hh * C + c];
      ss += v * a_s[hh * C + c];
      sd += v * a_d[hh * C + c];
    }
    als[(size_t)n * H + hh] = ss;
    ald[(size_t)n * H + hh] = sd;
  }
}

__device__ __forceinline__ void edge_endpoints(const int* ei, int idx, int E, int* s, int* d) {
  if (idx < E) { *s = ei[idx]; *d = ei[E + idx]; }
  else         { *s = *d = idx - E; }        // self loop
}

__global__ void edge_max_kernel(const int* __restrict__ ei, const float* __restrict__ als,
                                const float* __restrict__ ald, float* __restrict__ emax,
                                int E, int N, int H) {
  int idx = blockIdx.x * blockDim.x + threadIdx.x;
  if (idx >= E + N) return;
  int s, d; edge_endpoints(ei, idx, E, &s, &d);
  for (int hh = 0; hh < H; ++hh) {
    float e = als[(size_t)s * H + hh] + ald[(size_t)d * H + hh];
    e = (e > 0.f) ? e : 0.2f * e;           // leaky relu 0.2
    atomicMaxFloat(&emax[(size_t)d * H + hh], e);
  }
}

__global__ void edge_expsum_kernel(const int* __restrict__ ei, const float* __restrict__ als,
                                   const float* __restrict__ ald, const float* __restrict__ emax,
                                   float* __restrict__ den, int E, int N, int H) {
  int idx = blockIdx.x * blockDim.x + threadIdx.x;
  if (idx >= E + N) return;
  int s, d; edge_endpoints(ei, idx, E, &s, &d);
  for (int hh = 0; hh < H; ++hh) {
    float e = als[(size_t)s * H + hh] + ald[(size_t)d * H + hh];
    e = (e > 0.f) ? e : 0.2f * e;
    atomicAdd(&den[(size_t)d * H + hh], expf(e - emax[(size_t)d * H + hh]));
  }
}

__global__ void edge_agg_kernel(const int* __restrict__ ei, const float* __restrict__ h,
                                const float* __restrict__ als, const float* __restrict__ ald,
                                const float* __restrict__ emax, const float* __restrict__ den,
                                float* __restrict__ out, int E, int N, int H) {
  int idx = blockIdx.x * blockDim.x + threadIdx.x;
  if (idx >= E + N) return;
  int s, d; edge_endpoints(ei, idx, E, &s, &d);
  int C = HIDF / H;
  for (int hh = 0; hh < H; ++hh) {
    float e = als[(size_t)s * H + hh] + ald[(size_t)d * H + hh];
    e = (e > 0.f) ? e : 0.2f * e;
    float ex = expf(e - emax[(size_t)d * H + hh]);
    float alpha = ex / (den[(size_t)d * H + hh] + 1e-16f);
    const float* hs = h + (size_t)s * HIDF + hh * C;
    float* od = out + (size_t)d * HIDF + hh * C;
    for (int c = 0; c < C; ++c) atomicAdd(&od[c], hs[c] * alpha);
  }
}

// ---------------------------------------------------------------------------
// BatchNorm stats (population var) with bias folded in; one block per feature.
// ---------------------------------------------------------------------------
__global__ void bn_stats_kernel(const float* __restrict__ hagg, const float* __restrict__ bias,
                                float* __restrict__ mean, float* __restrict__ rstd, int N) {
  int f = blockIdx.x;
  int t = threadIdx.x;
  __shared__ float s1[256], s2[256];
  float a = 0.f, q = 0.f;
  float bf = bias[f];
  for (int n = t; n < N; n += 256) {
    float v = hagg[(size_t)n * HIDF + f] + bf;
    a += v; q += v * v;
  }
  s1[t] = a; s2[t] = q;
  __syncthreads();
  for (int s = 128; s > 0; s >>= 1) {
    if (t < s) { s1[t] += s1[t + s]; s2[t] += s2[t + s]; }
    __syncthreads();
  }
  if (t == 0) {
    float m = s1[0] / (float)N;
    float v = s2[0] / (float)N - m * m;
    mean[f] = m;
    rstd[f] = rsqrtf(v + 1e-5f);
  }
}

__global__ void bn_apply_relu_kernel(float* __restrict__ hagg, const float* __restrict__ bias,
                                     const float* __restrict__ g, const float* __restrict__ be,
                                     const float* __restrict__ mean, const float* __restrict__ rstd,
                                     int total) {
  int i = blockIdx.x * blockDim.x + threadIdx.x;
  if (i >= total) return;
  int f = i & (HIDF - 1);
  float v = (hagg[i] + bias[f] - mean[f]) * rstd[f] * g[f] + be[f];
  hagg[i] = fmaxf(v, 0.f);
}

// ---------------------------------------------------------------------------
// global mean pool + MLP head
// ---------------------------------------------------------------------------
__global__ void pool_count_kernel(const int* __restrict__ batch, float* __restrict__ cnt, int N) {
  int n = blockIdx.x * blockDim.x + threadIdx.x;
  if (n < N) atomicAdd(&cnt[batch[n]], 1.0f);
}

__global__ void pool_acc_kernel(const float* __restrict__ h, const int* __restrict__ batch,
                                float* __restrict__ pooled, int N) {
  int i = blockIdx.x * blockDim.x + threadIdx.x;
  if (i >= N * HIDF) return;
  int n = i >> 7;          // /128
  int f = i & (HIDF - 1);
  atomicAdd(&pooled[(size_t)batch[n] * HIDF + f], h[i]);
}

__global__ void mlp_kernel(const float* __restrict__ pooled, const float* __restrict__ cnt,
                           const float* __restrict__ fw1, const float* __restrict__ fb1,
                           const float* __restrict__ fw2, const float* __restrict__ fb2,
                           float* __restrict__ out) {
  int gidx = blockIdx.x;
  int j = threadIdx.x;     // 64 threads
  __shared__ float zsh[64];
  float inv = 1.0f / fmaxf(cnt[gidx], 1.0f);
  float acc = fb1[j];
  const float* pg = pooled + (size_t)gidx * HIDF;
  for (int k = 0; k < HIDF; ++k) acc += pg[k] * inv * fw1[k * 64 + j];
  zsh[j] = fmaxf(acc, 0.f) * fw2[j];
  __syncthreads();
  for (int s = 32; s > 0; s >>= 1) {
    if (j < s) zsh[j] += zsh[j + s];
    __syncthreads();
  }
  if (j == 0) out[gidx] = zsh[0] + fb2[0];
}

// ---------------------------------------------------------------------------
// launch
// ---------------------------------------------------------------------------
extern "C" void kernel_launch(void* const* d_in, const int* in_sizes, int n_in,
                              void* d_out, int out_size, void* d_ws, size_t ws_size,
                              hipStream_t stream) {
  const float* x     = (const float*)d_in[0];
  const int*   ei    = (const int*)d_in[1];
  const int*   batch = (const int*)d_in[2];
  const float* W1  = (const float*)d_in[3];
  const float* as1 = (const float*)d_in[4];
  const float* ad1 = (const float*)d_in[5];
  const float* b1  = (const float*)d_in[6];
  const float* g1  = (const float*)d_in[7];
  const float* be1 = (const float*)d_in[8];
  const float* W2  = (const float*)d_in[9];
  const float* as2 = (const float*)d_in[10];
  const float* ad2 = (const float*)d_in[11];
  const float* b2  = (const float*)d_in[12];
  const float* g2  = (const float*)d_in[13];
  const float* be2 = (const float*)d_in[14];
  const float* W3  = (const float*)d_in[15];
  const float* as3 = (const float*)d_in[16];
  const float* ad3 = (const float*)d_in[17];
  const float* b3  = (const float*)d_in[18];
  const float* g3  = (const float*)d_in[19];
  const float* be3 = (const float*)d_in[20];
  const float* fw1 = (const float*)d_in[21];
  const float* fb1 = (const float*)d_in[22];
  const float* fw2 = (const float*)d_in[23];
  const float* fb2 = (const float*)d_in[24];

  const int N   = in_sizes[2];
  const int E   = in_sizes[1] / 2;
  const int FIN = in_sizes[0] / N;
  const int KP  = ((FIN + 3) / 4) * 4;     // padded K for layer 1 (22 -> 24)
  const int NG  = out_size;
  float* out = (float*)d_out;

  // workspace layout (floats)
  float* buf0   = (float*)d_ws;                      // [N,128] gemm output h
  float* buf1   = buf0 + (size_t)N * HIDF;           // [N,128] agg / layer output
  float* als    = buf1 + (size_t)N * HIDF;           // [N,4]
  float* ald    = als + (size_t)N * 4;               // [N,4]
  float* emax   = ald + (size_t)N * 4;               // [N,4]
  float* den    = emax + (size_t)N * 4;              // [N,4]
  float* mean   = den + (size_t)N * 4;               // [128]
  float* rstd   = mean + HIDF;                       // [128]
  float* pooled = rstd + HIDF;                       // [NG,128]
  float* cnt    = pooled + (size_t)NG * HIDF;        // [NG]
  float* xpad   = cnt + NG;                          // [N,KP]

  const int Etot = E + N;
  const int nh_total = N * HIDF;

  auto run_layer = [&](const float* hin, int K, const float* W, int H,
                       const float* a_s, const float* a_d, const float* bias,
                       const float* gamma, const float* beta, float* hout) {
    int rowTiles = (N + 15) / 16;
    int tiles = rowTiles * 2;              // 2 col-groups of 64
    gemm_wmma<<<(tiles + 3) / 4, 128, 0, stream>>>(hin, W, buf0, N, K, tiles);
    scores_kernel<<<(N + 255) / 256, 256, 0, stream>>>(buf0, a_s, a_d, als, ald, N, H);
    fill_kernel<<<(N * H + 255) / 256, 256, 0, stream>>>(emax, N * H, -3.402823466e38f);
    fill_kernel<<<(N * H + 255) / 256, 256, 0, stream>>>(den, N * H, 0.f);
    fill_kernel<<<(nh_total + 255) / 256, 256, 0, stream>>>(hout, nh_total, 0.f);
    edge_max_kernel<<<(Etot + 255) / 256, 256, 0, stream>>>(ei, als, ald, emax, E, N, H);
    edge_expsum_kernel<<<(Etot + 255) / 256, 256, 0, stream>>>(ei, als, ald, emax, den, E, N, H);
    edge_agg_kernel<<<(Etot + 255) / 256, 256, 0, stream>>>(ei, buf0, als, ald, emax, den, hout, E, N, H);
    bn_stats_kernel<<<HIDF, 256, 0, stream>>>(hout, bias, mean, rstd, N);
    bn_apply_relu_kernel<<<(nh_total + 255) / 256, 256, 0, stream>>>(hout, bias, gamma, beta, mean, rstd, nh_total);
  };

  // Layer 1: pad K 22 -> 24 so the WMMA GEMM runs guard-free. Weight rows
  // 22..23 of the padded A are multiplied by whatever lies past W1 — avoid
  // that by padding A with zeros (done) so those products are exactly 0 only
  // if the A side is zero, which it is.
  pad_kernel<<<((N * KP) + 255) / 256, 256, 0, stream>>>(x, xpad, N, FIN, KP);
  // NOTE: gemm reads W[(k)*128 + ...] for k < KP; for k in [FIN,KP) the A
  // values are 0 so out-of-range W reads would still be wrong to perform.
  // W1 has FIN*128 = 2816 elements; reading rows 22..23 touches memory past
  // the logical end of W1. To stay safe, gemm uses K=KP but A==0 there, and
  // we clamp the W row index inside the hot loop is avoided by instead
  // padding W into workspace as well? Simpler: W reads for k<KP are within
  // the same allocation only if padded. So we pad W1 too:
  //   (handled below via wpad buffer)
  float* wpad = xpad + (size_t)N * KP;               // [KP,128]
  fill_kernel<<<((KP * HIDF) + 255) / 256, 256, 0, stream>>>(wpad, KP * HIDF, 0.f);
  // copy W1 rows [0,FIN) into wpad
  {
    int total = FIN * HIDF;
    // reuse pad_kernel semantics via a plain copy: pad_kernel with FIN==KP
    // would not zero-extend rows, so do a dedicated copy using fill+memcpy
    // pattern: simple elementwise copy kernel below (lambda not allowed,
    // use hipMemcpyAsync d2d which is graph-capture safe).
    hipMemcpyAsync(wpad, W1, (size_t)total * sizeof(float),
                   hipMemcpyDeviceToDevice, stream);
  }

  run_layer(xpad, KP,   wpad, 4, as1, ad1, b1, g1, be1, buf1);
  run_layer(buf1, HIDF, W2,   4, as2, ad2, b2, g2, be2, buf1);
  run_layer(buf1, HIDF, W3,   1, as3, ad3, b3, g3, be3, buf1);

  fill_kernel<<<(NG * HIDF + 255) / 256, 256, 0, stream>>>(pooled, NG * HIDF, 0.f);
  fill_kernel<<<(NG + 255) / 256, 256, 0, stream>>>(cnt, NG, 0.f);
  pool_count_kernel<<<(N + 255) / 256, 256, 0, stream>>>(batch, cnt, N);
  pool_acc_kernel<<<(nh_total + 255) / 256, 256, 0, stream>>>(buf1, batch, pooled, N);
  mlp_kernel<<<NG, 64, 0, stream>>>(pooled, cnt, fw1, fb1, fw2, fb2, out);
}